// RQSBijector_79104707658008
// MI455X (gfx1250) — compile-verified
//
#include <hip/hip_runtime.h>
#include <math.h>

// 4-wide float vector -> global_load_b128 / global_store_b128
typedef float v4f __attribute__((ext_vector_type(4)));

#define K_BINS 32
// table layout in d_ws (floats):
//  [0..32]   x_pos (33)
//  [33..65]  y_pos (33)
//  [66..98]  slopes (33)
//  [99]      log(slopes[0])
//  [100]     log(slopes[32])

__global__ void rqs_setup_kernel(const float* __restrict__ p, float* __restrict__ tab) {
    if (threadIdx.x != 0 || blockIdx.x != 0) return;
    const float MINB = 1.0e-4f;   // MIN_BIN_SIZE
    const float MINS = 1.0e-4f;   // MIN_SLOPE
    float w[K_BINS], h[K_BINS];
    float mw = -3.4e38f, mh = -3.4e38f;
    for (int i = 0; i < K_BINS; ++i) { w[i] = p[i];          mw = fmaxf(mw, w[i]); }
    for (int i = 0; i < K_BINS; ++i) { h[i] = p[K_BINS + i]; mh = fmaxf(mh, h[i]); }
    float sw = 0.f, sh = 0.f;
    for (int i = 0; i < K_BINS; ++i) { w[i] = expf(w[i] - mw); sw += w[i]; }
    for (int i = 0; i < K_BINS; ++i) { h[i] = expf(h[i] - mh); sh += h[i]; }
    const float scale = 10.0f - (float)K_BINS * MINB;  // (RANGE_MAX-RANGE_MIN) - K*MIN_BIN_SIZE
    const float isw = 1.0f / sw, ish = 1.0f / sh;
    float cx = -5.0f, cy = -5.0f;
    tab[0]  = -5.0f;
    tab[33] = -5.0f;
    for (int i = 0; i < K_BINS; ++i) {
        cx += w[i] * isw * scale + MINB;
        cy += h[i] * ish * scale + MINB;
        tab[1 + i]  = cx;
        tab[34 + i] = cy;
    }
    const float off = logf(expf(1.0f - MINS) - 1.0f);  // softplus offset
    for (int i = 0; i <= K_BINS; ++i) {
        float z  = p[2 * K_BINS + i] + off;
        float sp = (z > 20.0f) ? z : log1pf(expf(z));  // softplus
        tab[66 + i] = sp + MINS;
    }
    tab[99]  = logf(tab[66]);
    tab[100] = logf(tab[66 + K_BINS]);
}

__device__ __forceinline__ void rqs_eval(float x,
                                         const float* __restrict__ sx,
                                         const float* __restrict__ sy,
                                         const float* __restrict__ ss,
                                         float s_left, float s_right,
                                         float ldl, float ldr,
                                         float& yo, float& lo) {
    // searchsorted(x_pos[1:31], x, 'right') clipped to [0,31]:
    // largest idx in [0,31] with x_pos[idx] <= x (branchless binary search in LDS)
    int idx = (sx[16] <= x) ? 16 : 0;
    if (sx[idx + 8] <= x) idx += 8;
    if (sx[idx + 4] <= x) idx += 4;
    if (sx[idx + 2] <= x) idx += 2;
    if (sx[idx + 1] <= x) idx += 1;

    float xk  = sx[idx], xk1 = sx[idx + 1];
    float yk  = sy[idx], yk1 = sy[idx + 1];
    float dk  = ss[idx], dk1 = ss[idx + 1];

    float width  = xk1 - xk;
    float height = yk1 - yk;
    float inv_w  = __builtin_amdgcn_rcpf(width);            // v_rcp_f32 (width >= 1e-4)
    float xi     = (x - xk) * inv_w;
    xi           = fminf(fmaxf(xi, 0.0f), 1.0f);            // -> v_med3
    float s      = height * inv_w;
    float xim    = 1.0f - xi;
    float xxm    = xi * xim;

    // num = s*xi^2 + dk*xi*(1-xi) = xi * (s*xi + dk*(1-xi))
    float num = xi * fmaf(s, xi, dk * xim);
    // den = s + (dk1 + dk - 2s) * xi*(1-xi)
    float den = fmaf(fmaf(-2.0f, s, dk1 + dk), xxm, s);
    float inv_den = __builtin_amdgcn_rcpf(den);
    float y_sp = fmaf(height * num, inv_den, yk);

    // numd = s^2 * (dk1*xi^2 + 2s*xi(1-xi) + dk*(1-xi)^2)
    float t    = fmaf(dk1 * xi, xi, fmaf(s + s, xxm, dk * xim * xim));
    float sid  = s * inv_den;
    float deriv = (sid * sid) * t;                           // = s^2 * t / den^2
    // deriv is a positive normal: raw v_log_f32 (log2) * ln2, skip subnormal fixup
    float ld_sp = __builtin_amdgcn_logf(deriv) * 0.69314718055994531f;

    bool below = x < -5.0f;
    bool above = x >  5.0f;
    float y = below ? fmaf(x + 5.0f, s_left, -5.0f)
            : above ? fmaf(x - 5.0f, s_right, 5.0f)
                    : y_sp;
    float l = below ? ldl : (above ? ldr : ld_sp);
    yo = y; lo = l;
}

__global__ __launch_bounds__(256) void rqs_main_kernel(
    const v4f* __restrict__ x4, const float* __restrict__ tab,
    float* __restrict__ out_y, float* __restrict__ out_ld, int n4) {
    __shared__ float sx[33];
    __shared__ float sy[33];
    __shared__ float ss[33];
    __shared__ float sex[2];
    const int tid = threadIdx.x;
    if (tid < 101) {
        float v = tab[tid];
        if      (tid < 33)  sx[tid]       = v;
        else if (tid < 66)  sy[tid - 33]  = v;
        else if (tid < 99)  ss[tid - 66]  = v;
        else                sex[tid - 99] = v;
    }
    __syncthreads();
    const float s_left  = ss[0];
    const float s_right = ss[K_BINS];
    const float ldl = sex[0], ldr = sex[1];

    v4f* __restrict__ oy4 = (v4f*)out_y;
    v4f* __restrict__ ol4 = (v4f*)out_ld;

    const int stride = gridDim.x * blockDim.x;

    // two grid-stride streams per iteration: 8 independent element chains
    // -> more VOPD dual-issue, hides the dependent ds_load search chain
    int i = blockIdx.x * blockDim.x + tid;
    for (; i + stride < n4; i += 2 * stride) {
        const int j = i + stride;
        // streaming data (402 MB total > 192 MB L2): non-temporal hints (TH=NT)
        v4f xa = __builtin_nontemporal_load(&x4[i]);
        v4f xb = __builtin_nontemporal_load(&x4[j]);
        v4f ya, la, yb, lb;
#pragma unroll
        for (int c = 0; c < 4; ++c) {
            float yo, lo;
            rqs_eval(xa[c], sx, sy, ss, s_left, s_right, ldl, ldr, yo, lo);
            ya[c] = yo; la[c] = lo;
            rqs_eval(xb[c], sx, sy, ss, s_left, s_right, ldl, ldr, yo, lo);
            yb[c] = yo; lb[c] = lo;
        }
        __builtin_nontemporal_store(ya, &oy4[i]);
        __builtin_nontemporal_store(la, &ol4[i]);
        __builtin_nontemporal_store(yb, &oy4[j]);
        __builtin_nontemporal_store(lb, &ol4[j]);
    }
    if (i < n4) {
        v4f xa = __builtin_nontemporal_load(&x4[i]);
        v4f ya, la;
#pragma unroll
        for (int c = 0; c < 4; ++c) {
            float yo, lo;
            rqs_eval(xa[c], sx, sy, ss, s_left, s_right, ldl, ldr, yo, lo);
            ya[c] = yo; la[c] = lo;
        }
        __builtin_nontemporal_store(ya, &oy4[i]);
        __builtin_nontemporal_store(la, &ol4[i]);
    }
}

extern "C" void kernel_launch(void* const* d_in, const int* in_sizes, int n_in,
                              void* d_out, int out_size, void* d_ws, size_t ws_size,
                              hipStream_t stream) {
    const float* x      = (const float*)d_in[0];   // (8192, 4096) fp32
    const float* params = (const float*)d_in[1];   // (97,) fp32
    float* tab = (float*)d_ws;                     // 101 floats of scratch

    const int N  = in_sizes[0];
    const int n4 = N >> 2;                          // 4096 cols -> divisible by 4
    float* out_y  = (float*)d_out;                  // y first, logdet second
    float* out_ld = out_y + N;

    rqs_setup_kernel<<<1, 64, 0, stream>>>(params, tab);

    const int threads = 256;                        // 8 wave32 per block
    int blocks = 4096;                              // grid-stride, ~8 v4 per thread
    int needed = (n4 + 2 * threads - 1) / (2 * threads);
    if (blocks > needed) blocks = needed;
    if (blocks < 1) blocks = 1;
    rqs_main_kernel<<<blocks, threads, 0, stream>>>((const v4f*)x, tab, out_y, out_ld, n4);
}